// ElmoDecoder_14078902796954
// MI455X (gfx1250) — compile-verified
//
#include <hip/hip_runtime.h>
#include <hip/hip_bf16.h>
#include <math.h>

// Problem constants (reference: B=32, T=50, H=1024, D=512, V=50257)
#define Bb   32
#define Tt   50
#define Hh   1024
#define Dd   512
#define Vv   50257
#define G4   4096   // 4*H
#define MTOT 1600   // B*T

typedef __attribute__((ext_vector_type(2))) float v2f;
typedef __attribute__((ext_vector_type(8))) float v8f;

union V8 { v8f v; float f[8]; };

// CDNA5 dense fp32 matrix op: D(16x16,f32) = A(16x4,f32) x B(4x16,f32) + C
// 8-arg VOP3P form: (neg_a, A, neg_b, B, c_mod, C, reuse_a, reuse_b)
__device__ __forceinline__ v8f wmma_f32_16x16x4(v2f a, v2f b, v8f c) {
  return __builtin_amdgcn_wmma_f32_16x16x4_f32(false, a, false, b, (short)0, c,
                                               false, false);
}

// Load a 2-VGPR fp32 fragment from a [16 x 32] LDS tile (padded row stride 33).
// Per ISA layout (A 16x4 and B 4x16 mirror each other):
//   lanes 0-15 : element row = lane,      k = kk + {0,1}
//   lanes 16-31: element row = lane - 16, k = kk + {2,3}
__device__ __forceinline__ v2f frag_ld(const float* tile, int lane, int kk) {
  const int row = lane & 15;
  const int col = kk + ((lane >> 4) << 1);
  const float* p = tile + row * 33 + col;
  v2f f;
  f.x = p[0];
  f.y = p[1];
  return f;
}

// ---------------------------------------------------------------------------
// Kernel 1: xg[m, n] = sum_k emb[tok(m), k] * W_ih[n, k] + b_ih[n] + b_hh[n]
//   M = 1600 (b*T+t), N = 4096, K = 512.  Block = 128 thr (4 waves),
//   tile = 16(M) x 64(N), each wave owns a 16x16 WMMA tile.
// ---------------------------------------------------------------------------
__global__ __launch_bounds__(128)
void xg_gemm_kernel(const int* __restrict__ targets,
                    const float* __restrict__ emb,
                    const float* __restrict__ W_ih,
                    const float* __restrict__ b_ih,
                    const float* __restrict__ b_hh,
                    float* __restrict__ xg) {
  __shared__ float As[16 * 33];
  __shared__ float Ws[4][16 * 33];

  const int tid  = threadIdx.x;
  const int lane = tid & 31;
  const int w    = tid >> 5;                    // wave 0..3
  const int m0   = blockIdx.y << 4;             // M tile base
  const int n0   = (blockIdx.x << 6) + (w << 4);// this wave's N base

  v8f acc = {0.f, 0.f, 0.f, 0.f, 0.f, 0.f, 0.f, 0.f};

  // A staging: thread owns (row ar, 4 floats at ac). Gather token once.
  const int ar = tid >> 3;                      // 0..15
  const int ac = (tid & 7) << 2;                // 0..28
  const int am = m0 + ar;
  const int bb = am / Tt;
  const int tt = am % Tt;
  const int tok = (tt == 0) ? 1 : targets[bb * Tt + tt - 1];
  const float* arow = emb + (size_t)tok * Dd;

  // W staging: each wave stages its own 16 N-rows; lane owns 16 floats.
  const int wn = n0 + (lane >> 1);
  const int wk = (lane & 1) << 4;               // 0 or 16
  const float* wrow = W_ih + (size_t)wn * Dd + wk;
  float* wdst = Ws[w] + (lane >> 1) * 33 + wk;

  for (int k0 = 0; k0 < Dd; k0 += 32) {
    {
      const float4 va = *(const float4*)(arow + k0 + ac);
      float* d = As + ar * 33 + ac;
      d[0] = va.x; d[1] = va.y; d[2] = va.z; d[3] = va.w;
    }
    {
      const float* s = wrow + k0;
#pragma unroll
      for (int j = 0; j < 16; j += 4) {
        const float4 vw = *(const float4*)(s + j);
        wdst[j + 0] = vw.x; wdst[j + 1] = vw.y;
        wdst[j + 2] = vw.z; wdst[j + 3] = vw.w;
      }
    }
    if (k0 + 32 < Dd) {                         // global_prefetch_b8 next chunk
      __builtin_prefetch((const void*)(arow + k0 + 32 + ac), 0, 1);
      __builtin_prefetch((const void*)(wrow + k0 + 32), 0, 1);
    }
    __syncthreads();
#pragma unroll
    for (int kk = 0; kk < 32; kk += 4) {
      v2f af = frag_ld(As, lane, kk);
      v2f bf = frag_ld(Ws[w], lane, kk);
      acc = wmma_f32_16x16x4(af, bf, acc);
    }
    __syncthreads();
  }

  // Epilogue: C layout — VGPR r, lanes 0-15 -> M=r, lanes 16-31 -> M=r+8.
  const int col   = n0 + (lane & 15);
  const float bias = b_ih[col] + b_hh[col];
  const int rbase = m0 + ((lane >> 4) << 3);
  V8 res; res.v = acc;
#pragma unroll
  for (int r = 0; r < 8; ++r)
    xg[(size_t)(rbase + r) * G4 + col] = res.f[r] + bias;
}

// ---------------------------------------------------------------------------
// Kernel 2: one LSTM time step (launched 50x sequentially).
//   g = h_prev @ W_hh^T + xg[:,t,:], then gates -> c,h update.
//   Block = 256 thr (8 waves), grid = H/16; block covers a 16-wide h slice:
//   wave w -> gate q = w>>1, batch half mi = w&1. Gate tiles staged in LDS,
//   fused elementwise epilogue.
// ---------------------------------------------------------------------------
__global__ __launch_bounds__(256)
void lstm_step_kernel(const float* __restrict__ Abase, int rowStride, int rowOff,
                      const float* __restrict__ W_hh,
                      const float* __restrict__ xg,
                      float* __restrict__ hs,
                      float* __restrict__ cs,
                      int t) {
  __shared__ float As[32 * 33];
  __shared__ float Ws[4][16 * 33];
  __shared__ float Gs[32 * 68];   // stride 68 -> conflict-free half-wave writes

  const int tid  = threadIdx.x;
  const int lane = tid & 31;
  const int w    = tid >> 5;      // 0..7
  const int q    = w >> 1;        // gate 0..3 (i,f,g,o)
  const int mi   = w & 1;         // batch-half
  const int j0   = blockIdx.x << 4;

  v8f acc = {0.f, 0.f, 0.f, 0.f, 0.f, 0.f, 0.f, 0.f};

  const int ar = tid >> 3;                     // 0..31 (batch row)
  const int ac = (tid & 7) << 2;
  const float* arow = Abase + (size_t)ar * rowStride + rowOff;

  const int tq  = tid >> 6;                    // 0..3 (which gate tile)
  const int sub = tid & 63;
  const int wr  = sub >> 2;                    // 0..15
  const int wc  = (sub & 3) << 3;              // 0,8,16,24
  const float* wrow = W_hh + (size_t)(tq * Hh + j0 + wr) * Hh + wc;
  float* wdst = Ws[tq] + wr * 33 + wc;

  for (int k0 = 0; k0 < Hh; k0 += 32) {
    {
      const float4 va = *(const float4*)(arow + k0 + ac);
      float* d = As + ar * 33 + ac;
      d[0] = va.x; d[1] = va.y; d[2] = va.z; d[3] = va.w;
    }
    {
      const float* s = wrow + k0;
      const float4 v0 = *(const float4*)(s);
      const float4 v1 = *(const float4*)(s + 4);
      wdst[0] = v0.x; wdst[1] = v0.y; wdst[2] = v0.z; wdst[3] = v0.w;
      wdst[4] = v1.x; wdst[5] = v1.y; wdst[6] = v1.z; wdst[7] = v1.w;
    }
    if (k0 + 32 < Hh) {
      __builtin_prefetch((const void*)(arow + k0 + 32 + ac), 0, 1);
      __builtin_prefetch((const void*)(wrow + k0 + 32), 0, 1);
    }
    __syncthreads();
    const float* atile = As + (mi << 4) * 33;
#pragma unroll
    for (int kk = 0; kk < 32; kk += 4) {
      v2f af = frag_ld(atile, lane, kk);
      v2f bf = frag_ld(Ws[q], lane, kk);
      acc = wmma_f32_16x16x4(af, bf, acc);
    }
    __syncthreads();
  }

  // Stage gate tile to LDS: Gs[b][q*16 + jc]
  {
    V8 res; res.v = acc;
    const int jc = lane & 15;
    const int rb = (mi << 4) + ((lane >> 4) << 3);
#pragma unroll
    for (int r = 0; r < 8; ++r)
      Gs[(rb + r) * 68 + (q << 4) + jc] = res.f[r];
  }
  __syncthreads();

  // Fused gate math: 32 batch x 16 cols = 512 elems / 256 threads
#pragma unroll
  for (int e = 0; e < 2; ++e) {
    const int idx = tid + (e << 8);
    const int bm  = idx >> 4;                  // batch
    const int jc  = idx & 15;
    const int j   = j0 + jc;
    const size_t xb = ((size_t)bm * Tt + t) * G4 + j;
    const float gi_ = Gs[bm * 68 +  0 + jc] + xg[xb + 0 * Hh];
    const float gf_ = Gs[bm * 68 + 16 + jc] + xg[xb + 1 * Hh];
    const float gg_ = Gs[bm * 68 + 32 + jc] + xg[xb + 2 * Hh];
    const float go_ = Gs[bm * 68 + 48 + jc] + xg[xb + 3 * Hh];
    const float i_ = 1.0f / (1.0f + expf(-gi_));
    const float f_ = 1.0f / (1.0f + expf(-gf_));
    const float g_ = tanhf(gg_);
    const float o_ = 1.0f / (1.0f + expf(-go_));
    const float cold = cs[bm * Hh + j];
    const float cnew = f_ * cold + i_ * g_;
    cs[bm * Hh + j] = cnew;
    hs[((size_t)bm * Tt + t) * Hh + j] = o_ * tanhf(cnew);
  }
}

// ---------------------------------------------------------------------------
// Kernel 3: logits[m, v] = sum_k hs[m, k] * W_proj[v, k] + b_proj[v]
//   M = 1600, N = 50257 (guarded), K = 1024.
//   Block = 128 thr (4 waves), block tile = 16(M) x 256(N); each wave owns a
//   16x64 tile = 4 accumulators sharing every A fragment (DS/WMMA: 2 -> 1.25).
//   Tile ti (0..15) covers cols nblk + ti*16; wave w owns ti = {w, w+4, w+8, w+12}.
// ---------------------------------------------------------------------------
__global__ __launch_bounds__(128)
void proj_gemm_kernel(const float* __restrict__ hs,
                      const float* __restrict__ W_proj,
                      const float* __restrict__ b_proj,
                      float* __restrict__ logits) {
  __shared__ float As[16 * 33];
  __shared__ float Ws[16][16 * 33];

  const int tid  = threadIdx.x;
  const int lane = tid & 31;
  const int w    = tid >> 5;                  // wave 0..3
  const int m0   = blockIdx.y << 4;
  const int nblk = blockIdx.x << 8;           // 256 cols per block

  v8f acc[4];
#pragma unroll
  for (int s = 0; s < 4; ++s)
    acc[s] = (v8f){0.f, 0.f, 0.f, 0.f, 0.f, 0.f, 0.f, 0.f};

  const int ar = tid >> 3;
  const int ac = (tid & 7) << 2;
  const float* arow = hs + (size_t)(m0 + ar) * Hh;

  // W staging: wave w stages its 4 tiles; lane owns 16 floats per tile.
  const int nr = lane >> 1;                   // 0..15
  const int kb = (lane & 1) << 4;             // 0 or 16
  const float* wrow[4];
  float*       wdst[4];
  bool         wvalid[4];
#pragma unroll
  for (int s = 0; s < 4; ++s) {
    const int ti = w + (s << 2);
    const int wn = nblk + (ti << 4) + nr;
    wvalid[s] = (wn < Vv);
    wrow[s]   = W_proj + (size_t)wn * Hh + kb;
    wdst[s]   = Ws[ti] + nr * 33 + kb;
  }

  for (int k0 = 0; k0 < Hh; k0 += 32) {
    {
      const float4 va = *(const float4*)(arow + k0 + ac);
      float* d = As + ar * 33 + ac;
      d[0] = va.x; d[1] = va.y; d[2] = va.z; d[3] = va.w;
    }
#pragma unroll
    for (int s = 0; s < 4; ++s) {
      if (wvalid[s]) {
        const float* src = wrow[s] + k0;
        const float4 v0 = *(const float4*)(src);
        const float4 v1 = *(const float4*)(src + 4);
        const float4 v2 = *(const float4*)(src + 8);
        const float4 v3 = *(const float4*)(src + 12);
        float* d = wdst[s];
        d[0]  = v0.x; d[1]  = v0.y; d[2]  = v0.z; d[3]  = v0.w;
        d[4]  = v1.x; d[5]  = v1.y; d[6]  = v1.z; d[7]  = v1.w;
        d[8]  = v2.x; d[9]  = v2.y; d[10] = v2.z; d[11] = v2.w;
        d[12] = v3.x; d[13] = v3.y; d[14] = v3.z; d[15] = v3.w;
      } else {
        float* d = wdst[s];
#pragma unroll
        for (int j = 0; j < 16; ++j) d[j] = 0.f;
      }
    }
    if (k0 + 32 < Hh) {
      __builtin_prefetch((const void*)(arow + k0 + 32 + ac), 0, 1);
#pragma unroll
      for (int s = 0; s < 4; ++s)
        if (wvalid[s])
          __builtin_prefetch((const void*)(wrow[s] + k0 + 32), 0, 1);
    }
    __syncthreads();
#pragma unroll
    for (int kk = 0; kk < 32; kk += 4) {
      const v2f af = frag_ld(As, lane, kk);     // shared by 4 WMMAs below
#pragma unroll
      for (int s = 0; s < 4; ++s) {
        const v2f bf = frag_ld(Ws[w + (s << 2)], lane, kk);
        acc[s] = wmma_f32_16x16x4(af, bf, acc[s]);
      }
    }
    __syncthreads();
  }

  // Epilogue: C layout — VGPR r, lanes 0-15 -> M=r, lanes 16-31 -> M=r+8.
  const int rbase = m0 + ((lane >> 4) << 3);
#pragma unroll
  for (int s = 0; s < 4; ++s) {
    const int col = nblk + ((w + (s << 2)) << 4) + (lane & 15);
    if (col < Vv) {
      const float bias = b_proj[col];
      V8 res; res.v = acc[s];
#pragma unroll
      for (int r = 0; r < 8; ++r)
        logits[(size_t)(rbase + r) * Vv + col] = res.f[r] + bias;
    }
  }
}

// ---------------------------------------------------------------------------
// Kernel 4: predictions[m] = argmax_v logits[m, v]  (first-index tie break)
// ---------------------------------------------------------------------------
__global__ __launch_bounds__(256)
void argmax_kernel(const float* __restrict__ logits, float* __restrict__ preds) {
  __shared__ float sval[256];
  __shared__ int   sidx[256];
  const int m = blockIdx.x;
  const float* row = logits + (size_t)m * Vv;

  float best = -INFINITY;
  int   bi   = 0x7fffffff;
  for (int v = threadIdx.x; v < Vv; v += 256) {
    const float x = row[v];
    if (x > best) { best = x; bi = v; }   // strided -> local first max kept
  }
  sval[threadIdx.x] = best;
  sidx[threadIdx.x] = bi;
  __syncthreads();
  for (int s = 128; s > 0; s >>= 1) {
    if (threadIdx.x < s) {
      const float ov = sval[threadIdx.x + s];
      const int   oi = sidx[threadIdx.x + s];
      if (ov > sval[threadIdx.x] ||
          (ov == sval[threadIdx.x] && oi < sidx[threadIdx.x])) {
        sval[threadIdx.x] = ov;
        sidx[threadIdx.x] = oi;
      }
    }
    __syncthreads();
  }
  if (threadIdx.x == 0) preds[m] = (float)sidx[0];
}

// ---------------------------------------------------------------------------
extern "C" void kernel_launch(void* const* d_in, const int* in_sizes, int n_in,
                              void* d_out, int out_size, void* d_ws, size_t ws_size,
                              hipStream_t stream) {
  (void)in_sizes; (void)n_in; (void)out_size; (void)ws_size;
  const float* inputs  = (const float*)d_in[0];   // [B,H]
  const int*   targets = (const int*)  d_in[1];   // [B,T]
  const float* emb     = (const float*)d_in[2];   // [V,D]
  const float* W_ih    = (const float*)d_in[3];   // [4H,D]
  const float* W_hh    = (const float*)d_in[4];   // [4H,H]
  const float* b_ih    = (const float*)d_in[5];   // [4H]
  const float* b_hh    = (const float*)d_in[6];   // [4H]
  const float* W_proj  = (const float*)d_in[7];   // [V,H]
  const float* b_proj  = (const float*)d_in[8];   // [V]

  float* logits = (float*)d_out;                      // [B*T, V]
  float* preds  = logits + (size_t)MTOT * Vv;         // [B*T]

  float* xg = (float*)d_ws;                           // [B*T, 4H]  26.2 MB
  float* hs = xg + (size_t)MTOT * G4;                 // [B*T, H]    6.6 MB
  float* cs = hs + (size_t)MTOT * Hh;                 // [B, H]      0.13 MB

  hipMemsetAsync(cs, 0, (size_t)Bb * Hh * sizeof(float), stream);

  // Phase 1: input-gate precompute (embedding gather fused into A staging)
  xg_gemm_kernel<<<dim3(G4 / 64, MTOT / 16), 128, 0, stream>>>(
      targets, emb, W_ih, b_ih, b_hh, xg);

  // Phase 2: sequential recurrence; W_hh (16 MB) stays hot in the 192 MB L2
  for (int t = 0; t < Tt; ++t) {
    const float* Ab = (t == 0) ? inputs : hs;
    const int rs = (t == 0) ? Hh : Tt * Hh;
    const int ro = (t == 0) ? 0 : (t - 1) * Hh;
    lstm_step_kernel<<<dim3(Hh / 16), 256, 0, stream>>>(
        Ab, rs, ro, W_hh, xg, hs, cs, t);
  }

  // Phase 3: vocabulary projection (dominant 164.7 GFLOP GEMM),
  //          16x64 per-wave tile -> 4x A-fragment reuse
  proj_gemm_kernel<<<dim3((Vv + 255) / 256, MTOT / 16), 128, 0, stream>>>(
      hs, W_proj, b_proj, logits);

  // Phase 4: argmax -> predictions
  argmax_kernel<<<dim3(MTOT), 256, 0, stream>>>(logits, preds);
}